// TPS_2448131359255
// MI455X (gfx1250) — compile-verified
//
#include <hip/hip_runtime.h>

typedef __attribute__((ext_vector_type(2))) float v2f;
typedef __attribute__((ext_vector_type(8))) float v8f;

#define TPS_EPS 1e-6f
#define LN2F    0.69314718055994530942f
constexpr int T       = 128;     // control points
constexpr int MM      = 132;     // T + DIM + 1
constexpr int NCOL    = 135;     // MM + 3 RHS columns
constexpr int ASTRIDE = 136;     // padded row stride of augmented system
constexpr int WSTRIDE = 132;     // padded row stride of transposed w in LDS
                                 // ((4m + t) % 64 -> conflict-free b64 loads)
constexpr int GP      = 64 * 64 * 64; // grid points per batch

// Raw hardware transcendentals: valid since r2 >= 1e-6 (no denorm/neg/inf).
// v_sqrt_f32 + v_log_f32(base2)*ln2, no libm fix-up sequences.
__device__ __forceinline__ float tps_u(float r2) {
    float r = __builtin_amdgcn_sqrtf(r2);
    return r2 * (LN2F * __builtin_amdgcn_logf(r + TPS_EPS));
}

// ---------------------------------------------------------------------------
// Kernel 1: assemble augmented system [A | rhs] per batch into d_ws
// ---------------------------------------------------------------------------
__global__ void tps_build(const float* __restrict__ pm, const float* __restrict__ pf,
                          const float* __restrict__ lmbda, float* __restrict__ Aaug) {
    int n = blockIdx.x;
    const float* pfn = pf + n * T * 3;
    const float* pmn = pm + n * T * 3;
    float lam = lmbda[n];
    float* A = Aaug + (size_t)n * MM * ASTRIDE;
    for (int idx = threadIdx.x; idx < MM * NCOL; idx += blockDim.x) {
        int r = idx / NCOL, c = idx % NCOL;
        float v = 0.0f;
        if (c < MM) {
            if (r < T && c < T) {
                float dx = pfn[r * 3 + 0] - pfn[c * 3 + 0];
                float dy = pfn[r * 3 + 1] - pfn[c * 3 + 1];
                float dz = pfn[r * 3 + 2] - pfn[c * 3 + 2];
                float r2 = dx * dx + dy * dy + dz * dz + TPS_EPS; // (sqrt(..+eps))^2
                v = tps_u(r2);
                if (r == c) v += lam;
            } else if (r < T) {               // P block
                int k = c - T;
                v = (k == 0) ? 1.0f : pfn[r * 3 + (k - 1)];
            } else if (c < T) {               // P^T block
                int k = r - T;
                v = (k == 0) ? 1.0f : pfn[c * 3 + (k - 1)];
            }
        } else {                              // RHS columns: points_m
            int d = c - MM;
            v = (r < T) ? pmn[r * 3 + d] : 0.0f;
        }
        A[r * ASTRIDE + c] = v;
    }
}

// ---------------------------------------------------------------------------
// Kernel 2: Gaussian elimination (partial pivoting) + back substitution.
// One block per batch. theta (MM x 3) written to d_ws.
// ---------------------------------------------------------------------------
__global__ void tps_solve(float* __restrict__ Aaug, float* __restrict__ theta) {
    int n = blockIdx.x;
    float* A = Aaug + (size_t)n * MM * ASTRIDE;
    __shared__ int   s_piv;
    __shared__ float s_f[MM];
    __shared__ float s_x[MM][3];

    for (int k = 0; k < MM; ++k) {
        if (threadIdx.x == 0) {
            int piv = k; float best = fabsf(A[k * ASTRIDE + k]);
            for (int r = k + 1; r < MM; ++r) {
                float v = fabsf(A[r * ASTRIDE + k]);
                if (v > best) { best = v; piv = r; }
            }
            s_piv = piv;
        }
        __syncthreads();
        int piv = s_piv;
        if (piv != k) {
            for (int c = k + threadIdx.x; c < NCOL; c += blockDim.x) {
                float t0 = A[k * ASTRIDE + c];
                A[k * ASTRIDE + c]   = A[piv * ASTRIDE + c];
                A[piv * ASTRIDE + c] = t0;
            }
        }
        __syncthreads();
        float pivv = A[k * ASTRIDE + k];
        for (int r = k + 1 + threadIdx.x; r < MM; r += blockDim.x)
            s_f[r] = A[r * ASTRIDE + k] / pivv;
        __syncthreads();
        int nr = MM - (k + 1), nc = NCOL - (k + 1);
        for (int idx = threadIdx.x; idx < nr * nc; idx += blockDim.x) {
            int r = k + 1 + idx / nc;
            int c = k + 1 + idx % nc;
            A[r * ASTRIDE + c] -= s_f[r] * A[k * ASTRIDE + c];
        }
        __syncthreads();
    }
    // Back substitution: 3 independent RHS columns, one thread each
    if (threadIdx.x < 3) {
        int d = threadIdx.x;
        for (int r = MM - 1; r >= 0; --r) {
            float s = A[r * ASTRIDE + MM + d];
            for (int c = r + 1; c < MM; ++c) s -= A[r * ASTRIDE + c] * s_x[c][d];
            s_x[r][d] = s / A[r * ASTRIDE + r];
        }
    }
    __syncthreads();
    for (int idx = threadIdx.x; idx < MM * 3; idx += blockDim.x)
        theta[(size_t)n * MM * 3 + idx] = s_x[idx / 3][idx % 3];
}

// ---------------------------------------------------------------------------
// Kernel 3: TPS evaluation. Each wave owns 16 grid points; U (16x128, fp32)
// is generated on the VALU (bare v_sqrt_f32 / v_log_f32 trans ops) and
// contracted with w (128x16-padded, pre-transposed in LDS) via 32 fully-
// unrolled chained V_WMMA_F32_16X16X4_F32 ops that co-execute with the
// transcendental chain. No EXEC divergence anywhere near the WMMAs.
// ---------------------------------------------------------------------------
__global__ void __launch_bounds__(256)
tps_eval(const float* __restrict__ pf, const float* __restrict__ theta,
         float* __restrict__ out) {
    __shared__ float4 s_pf[T];               // control point xyz + |pf|^2
    __shared__ float  s_wT[16 * WSTRIDE];    // w transposed+padded: wT[m][t]
    __shared__ float  s_a[4 * 3];            // affine rows (theta rows T..T+3)

    int n    = blockIdx.x >> 11;             // 2048 blocks per batch
    int tile = blockIdx.x & 2047;
    const float* pfn = pf + n * T * 3;
    const float* thn = theta + (size_t)n * MM * 3;

    __builtin_prefetch(pfn, 0, 3);           // global_prefetch_b8
    __builtin_prefetch(thn, 0, 3);

    if (threadIdx.x < T) {
        float x = pfn[threadIdx.x * 3 + 0];
        float y = pfn[threadIdx.x * 3 + 1];
        float z = pfn[threadIdx.x * 3 + 2];
        s_pf[threadIdx.x] = make_float4(x, y, z, x * x + y * y + z * z);
    }
    // Transposed, zero-padded B operand: s_wT[m * WSTRIDE + t] = w[t][m] (m<3)
    for (int idx = threadIdx.x; idx < 16 * T; idx += blockDim.x) {
        int mcol = idx >> 7;                 // 0..15
        int trow = idx & (T - 1);            // 0..127
        s_wT[mcol * WSTRIDE + trow] = (mcol < 3) ? thn[trow * 3 + mcol] : 0.0f;
    }
    if (threadIdx.x < 12) s_a[threadIdx.x] = thn[T * 3 + threadIdx.x];
    __syncthreads();

    int wave  = threadIdx.x >> 5;
    int lane  = threadIdx.x & 31;
    int hi    = lane >> 4;                   // half-wave: K-subgroup select
    int m     = lane & 15;                   // row (grid point) for A; col for B/C
    int pbase = tile * 128 + wave * 16;
    int p     = pbase + m;

    float gx = fmaf((float)(p & 63),         2.0f / 63.0f, -1.0f);
    float gy = fmaf((float)((p >> 6) & 63),  2.0f / 63.0f, -1.0f);
    float gz = fmaf((float)(p >> 12),        2.0f / 63.0f, -1.0f);
    float gdot = gx * gx + gy * gy + gz * gz;

    // Per-lane bases: A-frag covers K = c4 + 2*hi + {0,1}
    const float* pf_base = (const float*)&s_pf[2 * hi];         // + c4 per step
    const float* wT_base = &s_wT[m * WSTRIDE + 2 * hi];         // + c4 per step

    v8f acc = {};
#pragma unroll
    for (int c4 = 0; c4 < T; c4 += 4) {
        float4 q0 = ((const float4*)pf_base)[c4];
        float4 q1 = ((const float4*)pf_base)[c4 + 1];
        float r20 = gdot + q0.w - 2.0f * (gx * q0.x + gy * q0.y + gz * q0.z);
        float r21 = gdot + q1.w - 2.0f * (gx * q1.x + gy * q1.y + gz * q1.z);
        r20 = fmaxf(r20, 0.0f) + TPS_EPS;    // == r^2 after clamp+eps, >= 1e-6
        r21 = fmaxf(r21, 0.0f) + TPS_EPS;
        v2f a;                                // A fragment 16x4 f32
        a.x = tps_u(r20);
        a.y = tps_u(r21);
        v2f b = *(const v2f*)(wT_base + c4);  // B fragment 4x16 f32, one b64 load
        acc = __builtin_amdgcn_wmma_f32_16x16x4_f32(
            false, a, false, b, (short)0, acc, false, false);
    }

    // C/D layout: VGPR j -> row M=j (lanes 0-15) / M=j+8 (lanes 16-31), col N=lane&15
    if (m < 3) {
        int d = m;
        float a0 = s_a[0 * 3 + d], a1 = s_a[1 * 3 + d];
        float a2 = s_a[2 * 3 + d], a3 = s_a[3 * 3 + d];
#pragma unroll
        for (int j = 0; j < 8; ++j) {
            int ms = j + 8 * hi;
            int ps = pbase + ms;
            float xx = fmaf((float)(ps & 63),        2.0f / 63.0f, -1.0f);
            float yy = fmaf((float)((ps >> 6) & 63), 2.0f / 63.0f, -1.0f);
            float zz = fmaf((float)(ps >> 12),       2.0f / 63.0f, -1.0f);
            out[((size_t)n * GP + ps) * 3 + d] =
                acc[j] + a0 + xx * a1 + yy * a2 + zz * a3;
        }
    }
}

// ---------------------------------------------------------------------------
extern "C" void kernel_launch(void* const* d_in, const int* in_sizes, int n_in,
                              void* d_out, int out_size, void* d_ws, size_t ws_size,
                              hipStream_t stream) {
    const float* pm = (const float*)d_in[0];   // points_m (2,128,3)
    const float* pf = (const float*)d_in[1];   // points_f (2,128,3)
    const float* lm = (const float*)d_in[2];   // lmbda (2,)
    float* Aaug  = (float*)d_ws;               // 2 * 132 * 136 floats
    float* theta = Aaug + 2 * MM * ASTRIDE;    // 2 * 132 * 3 floats
    float* outp  = (float*)d_out;              // (2,64,64,64,3)

    tps_build<<<2, 256, 0, stream>>>(pm, pf, lm, Aaug);
    tps_solve<<<2, 256, 0, stream>>>(Aaug, theta);
    tps_eval <<<4096, 256, 0, stream>>>(pf, theta, outp);
}